// SVDHead_10024453668979
// MI455X (gfx1250) — compile-verified
//
#include <hip/hip_runtime.h>

// Problem constants (from reference): B=1024 batches, N=8192 points, 3D.
#define B_BATCH 1024
#define NPTS    8192
#define THREADS 256
#define CH      256          // points staged per chunk (one per thread)
#define KG      (CH / 4)     // 64 k-groups of 4 points per chunk
#define WAVES   (THREADS / 32)
#define KG_PER_WAVE (KG / WAVES) // 8 WMMAs per wave per chunk

typedef __attribute__((ext_vector_type(2))) float v2f;
typedef __attribute__((ext_vector_type(8))) float v8f;

// ---------------------------------------------------------------------------
// Kernel 1: streaming reduction.  One workgroup per batch.
// Computes C = L^T * R  (4x4, fp32) where per point n:
//   L[n,:] = (w, w*sx, w*sy, w*sz),  R[n,:] = (1, tx, ty, tz)
// via chained V_WMMA_F32_16X16X4_F32 (rows/cols 0..3 of the 16x16 tile used;
// the rest of the tile is kept exactly zero).
// C[0][0]=Sum w, C[1..3][0]=Sum w*s, C[0][1..3]=Sum w*t, C[1..3][1..3]=Sum w*s t^T
// ---------------------------------------------------------------------------
__global__ __launch_bounds__(THREADS)
void svd_reduce_kernel(const float* __restrict__ src,
                       const float* __restrict__ tpl,
                       const float* __restrict__ wgt,
                       float* __restrict__ part)
{
    // WMMA-layout staging tiles: tile[g][lane*2 + v] = operand VGPR v of lane.
    __shared__ float tileA[KG][64];
    __shared__ float tileB[KG][64];
    __shared__ float accLDS[16];

    const int tid  = threadIdx.x;
    const int lane = tid & 31;
    const int wv   = tid >> 5;
    const int b    = blockIdx.x;

    // Zero all LDS once: unused WMMA rows (m>=4) / cols (n>=4) stay exactly 0.
    for (int i = tid; i < KG * 64; i += THREADS) {
        (&tileA[0][0])[i] = 0.0f;
        (&tileB[0][0])[i] = 0.0f;
    }
    if (tid < 16) accLDS[tid] = 0.0f;

    const size_t base = (size_t)b * NPTS;
    const float* gs = src + base * 3;
    const float* gt = tpl + base * 3;
    const float* gw = wgt + base;

    v8f acc = {0.f, 0.f, 0.f, 0.f, 0.f, 0.f, 0.f, 0.f};

    // Staging slot for this thread's point within its k-group.
    // f32 A 16x4 layout: lane m (0-15) / lane 16+m hold K={0,1}/{2,3} in VGPRs 0/1.
    // Inverse map for point k (0..3), feature f (0..3):
    //   lds index = (f + 16*(k>>1))*2 + (k&1) = 32*(k>>1) + (k&1) + 2*f
    const int k        = tid & 3;
    const int gStage   = tid >> 2;
    const int slotBase = 32 * (k >> 1) + (k & 1);
    const int g0       = wv * KG_PER_WAVE;

    for (int c0 = 0; c0 < NPTS; c0 += CH) {
        const int p = c0 + tid;
        // Coalesced global loads (b96/b32 within contiguous 3 KB (s, t) and 1 KB (w) windows).
        const float w  = gw[p];
        const float s0 = gs[p * 3 + 0], s1 = gs[p * 3 + 1], s2 = gs[p * 3 + 2];
        const float t0 = gt[p * 3 + 0], t1 = gt[p * 3 + 1], t2 = gt[p * 3 + 2];

        if (c0 + CH < NPTS) {   // pull next chunk toward us (global_prefetch_b8)
            __builtin_prefetch(&gw[p + CH], 0, 1);
            __builtin_prefetch(&gs[(p + CH) * 3], 0, 1);
            __builtin_prefetch(&gt[(p + CH) * 3], 0, 1);
        }

        __syncthreads();  // previous chunk's WMMA reads are done
        float* ta = &tileA[gStage][slotBase];
        float* tb = &tileB[gStage][slotBase];
        ta[0] = w;        tb[0] = 1.0f;
        ta[2] = w * s0;   tb[2] = t0;
        ta[4] = w * s1;   tb[4] = t1;
        ta[6] = w * s2;   tb[6] = t2;
        __syncthreads();  // tiles ready

        // Each wave consumes its 8 k-groups with chained WMMAs (fp32, K=4).
#pragma unroll
        for (int gg = 0; gg < KG_PER_WAVE; ++gg) {
            const int g = g0 + gg;
            v2f a  = *(const v2f*)&tileA[g][lane * 2];
            v2f bb = *(const v2f*)&tileB[g][lane * 2];
            acc = __builtin_amdgcn_wmma_f32_16x16x4_f32(
                /*neg_a=*/false, a, /*neg_b=*/false, bb,
                /*c_mod=*/(short)0, acc, /*reuse_a=*/false, /*reuse_b=*/false);
        }
    }
    __syncthreads();

    // D layout: lane n (0-15), VGPR i => D[M=i][n].  Used block: i<4, n<4.
    if (lane < 4) {
        atomicAdd(&accLDS[0 * 4 + lane], acc[0]);
        atomicAdd(&accLDS[1 * 4 + lane], acc[1]);
        atomicAdd(&accLDS[2 * 4 + lane], acc[2]);
        atomicAdd(&accLDS[3 * 4 + lane], acc[3]);
    }
    __syncthreads();
    if (tid < 16) part[(size_t)b * 16 + tid] = accLDS[tid];
}

// ---------------------------------------------------------------------------
// Kernel 2: per-batch 3x3 Kabsch solve (one thread per batch).
// ---------------------------------------------------------------------------
__device__ inline float det3x3(const float M[3][3]) {
    return M[0][0] * (M[1][1] * M[2][2] - M[1][2] * M[2][1])
         - M[0][1] * (M[1][0] * M[2][2] - M[1][2] * M[2][0])
         + M[0][2] * (M[1][0] * M[2][1] - M[1][1] * M[2][0]);
}

__global__ __launch_bounds__(256)
void svd_solve_kernel(const float* __restrict__ part, float* __restrict__ out)
{
    const int b = blockIdx.x * blockDim.x + threadIdx.x;
    if (b >= B_BATCH) return;

    float C[4][4];
#pragma unroll
    for (int i = 0; i < 16; ++i) C[i >> 2][i & 3] = part[(size_t)b * 16 + i];

    const float W     = C[0][0];
    const float alpha = 1.0f / (W + 1e-8f);        // matches reference's +1e-8
    const float sig   = W * alpha;                 // sum of normalized weights
    float sc[3] = {C[1][0] * alpha, C[2][0] * alpha, C[3][0] * alpha};
    float tc[3] = {C[0][1] * alpha, C[0][2] * alpha, C[0][3] * alpha};

    // H = sum(w_hat s t^T) - (2 - sig) * sc tc^T   (exact centered covariance)
    float H[3][3];
#pragma unroll
    for (int d = 0; d < 3; ++d)
#pragma unroll
        for (int e = 0; e < 3; ++e)
            H[d][e] = alpha * C[d + 1][e + 1] - (2.0f - sig) * sc[d] * tc[e];

    // S = H^T H (symmetric), eigendecompose with cyclic Jacobi -> V, lambda.
    float S[3][3];
#pragma unroll
    for (int i = 0; i < 3; ++i)
#pragma unroll
        for (int j = 0; j < 3; ++j)
            S[i][j] = H[0][i] * H[0][j] + H[1][i] * H[1][j] + H[2][i] * H[2][j];

    float V[3][3] = {{1.f, 0.f, 0.f}, {0.f, 1.f, 0.f}, {0.f, 0.f, 1.f}};
    const int PQ[3][2] = {{0, 1}, {0, 2}, {1, 2}};
    for (int sweep = 0; sweep < 8; ++sweep) {
#pragma unroll
        for (int pi = 0; pi < 3; ++pi) {
            const int p = PQ[pi][0], q = PQ[pi][1];
            const float apq = S[p][q];
            if (fabsf(apq) > 1e-20f) {
                const float tau = (S[q][q] - S[p][p]) / (2.0f * apq);
                const float tt  = (tau >= 0.0f ? 1.0f : -1.0f) /
                                  (fabsf(tau) + sqrtf(1.0f + tau * tau));
                const float c = 1.0f / sqrtf(1.0f + tt * tt);
                const float s = tt * c;
#pragma unroll
                for (int r = 0; r < 3; ++r) {   // S = S * J
                    const float Srp = S[r][p], Srq = S[r][q];
                    S[r][p] = c * Srp - s * Srq;
                    S[r][q] = s * Srp + c * Srq;
                }
#pragma unroll
                for (int r = 0; r < 3; ++r) {   // S = J^T * S
                    const float Spr = S[p][r], Sqr = S[q][r];
                    S[p][r] = c * Spr - s * Sqr;
                    S[q][r] = s * Spr + c * Sqr;
                }
#pragma unroll
                for (int r = 0; r < 3; ++r) {   // V = V * J
                    const float Vrp = V[r][p], Vrq = V[r][q];
                    V[r][p] = c * Vrp - s * Vrq;
                    V[r][q] = s * Vrp + c * Vrq;
                }
            }
        }
    }

    // Sort eigenpairs descending (matches SVD singular-value order).
    float lam[3] = {S[0][0], S[1][1], S[2][2]};
#pragma unroll
    for (int pi = 0; pi < 3; ++pi) {
        const int i = PQ[pi][0], j = PQ[pi][1];
        if (lam[j] > lam[i]) {
            const float tl = lam[i]; lam[i] = lam[j]; lam[j] = tl;
#pragma unroll
            for (int r = 0; r < 3; ++r) {
                const float tv = V[r][i]; V[r][i] = V[r][j]; V[r][j] = tv;
            }
        }
    }

    // U columns: u_k = normalize(H v_k), with orthogonal completion.
    float Uc[3][3];  // Uc[k][r] = component r of U column k
#pragma unroll
    for (int kcol = 0; kcol < 2; ++kcol)
#pragma unroll
        for (int r = 0; r < 3; ++r)
            Uc[kcol][r] = H[r][0] * V[0][kcol] + H[r][1] * V[1][kcol] + H[r][2] * V[2][kcol];

    float n0 = sqrtf(Uc[0][0]*Uc[0][0] + Uc[0][1]*Uc[0][1] + Uc[0][2]*Uc[0][2]);
    if (n0 > 1e-12f) { const float r0 = 1.0f / n0; Uc[0][0]*=r0; Uc[0][1]*=r0; Uc[0][2]*=r0; }
    else { Uc[0][0] = 1.f; Uc[0][1] = 0.f; Uc[0][2] = 0.f; }

    const float d01 = Uc[0][0]*Uc[1][0] + Uc[0][1]*Uc[1][1] + Uc[0][2]*Uc[1][2];
    Uc[1][0] -= d01 * Uc[0][0]; Uc[1][1] -= d01 * Uc[0][1]; Uc[1][2] -= d01 * Uc[0][2];
    float n1 = sqrtf(Uc[1][0]*Uc[1][0] + Uc[1][1]*Uc[1][1] + Uc[1][2]*Uc[1][2]);
    if (n1 > 1e-12f) { const float r1 = 1.0f / n1; Uc[1][0]*=r1; Uc[1][1]*=r1; Uc[1][2]*=r1; }
    else {
        const float ax = (fabsf(Uc[0][0]) < 0.9f) ? 1.f : 0.f;
        const float ay = 1.f - ax;
        float cx = ay * Uc[0][2] - 0.f * Uc[0][1];
        float cy = 0.f * Uc[0][0] - ax * Uc[0][2];
        float cz = ax * Uc[0][1] - ay * Uc[0][0];
        const float cn = 1.0f / sqrtf(cx*cx + cy*cy + cz*cz + 1e-30f);
        Uc[1][0] = cx * cn; Uc[1][1] = cy * cn; Uc[1][2] = cz * cn;
    }
    Uc[2][0] = Uc[0][1]*Uc[1][2] - Uc[0][2]*Uc[1][1];
    Uc[2][1] = Uc[0][2]*Uc[1][0] - Uc[0][0]*Uc[1][2];
    Uc[2][2] = Uc[0][0]*Uc[1][1] - Uc[0][1]*Uc[1][0];

    // R_pos = V U^T ;  flip last column of V if det <= 0.
    float R[3][3];
#pragma unroll
    for (int i = 0; i < 3; ++i)
#pragma unroll
        for (int j = 0; j < 3; ++j)
            R[i][j] = V[i][0]*Uc[0][j] + V[i][1]*Uc[1][j] + V[i][2]*Uc[2][j];

    if (det3x3(R) <= 0.0f) {
#pragma unroll
        for (int i = 0; i < 3; ++i)
#pragma unroll
            for (int j = 0; j < 3; ++j)
                R[i][j] -= 2.0f * V[i][2] * Uc[2][j];
    }

    // t = tc - R sc
    float tvec[3];
#pragma unroll
    for (int i = 0; i < 3; ++i)
        tvec[i] = tc[i] - (R[i][0]*sc[0] + R[i][1]*sc[1] + R[i][2]*sc[2]);

    float* Rout = out;                       // [B,3,3] flat
    float* tout = out + (size_t)B_BATCH * 9; // [B,3] flat
#pragma unroll
    for (int i = 0; i < 3; ++i)
#pragma unroll
        for (int j = 0; j < 3; ++j)
            Rout[(size_t)b * 9 + i * 3 + j] = R[i][j];
#pragma unroll
    for (int i = 0; i < 3; ++i)
        tout[(size_t)b * 3 + i] = tvec[i];
}

// ---------------------------------------------------------------------------
extern "C" void kernel_launch(void* const* d_in, const int* in_sizes, int n_in,
                              void* d_out, int out_size, void* d_ws, size_t ws_size,
                              hipStream_t stream)
{
    const float* src = (const float*)d_in[0];  // [B,N,3]
    const float* tpl = (const float*)d_in[1];  // [B,N,3]
    const float* wgt = (const float*)d_in[2];  // [B,N]
    float* out  = (float*)d_out;               // R flat (B*9) then t flat (B*3)
    float* part = (float*)d_ws;                // B*16 floats of raw sums

    svd_reduce_kernel<<<B_BATCH, THREADS, 0, stream>>>(src, tpl, wgt, part);
    svd_solve_kernel<<<(B_BATCH + 255) / 256, 256, 0, stream>>>(part, out);
}